// MTM_Net_19310172963218
// MI455X (gfx1250) — compile-verified
//
#include <hip/hip_runtime.h>
#include <hip/hip_bf16.h>
#include <math.h>
#include <stdint.h>

// ---- model dims (match reference) ----
#define NUM_LAYERS 2
#define D_MODEL    512
#define D_INNER    1024
#define D_STATE    16
#define D_CONV     4
#define DT_RANK    32
#define BATCH      32
#define SEQ        128
#define ROWS       (BATCH * SEQ)        // 4096 token rows

#define KC 16                           // K-chunk per pipeline stage
#define PS 20                           // padded LDS row stride (floats); 16 rows*20 mod 64 banks all distinct

typedef __attribute__((ext_vector_type(2))) float v2f;
typedef __attribute__((ext_vector_type(8))) float v8f;

// ---------------------------------------------------------------------------
// f32 WMMA GEMM with async-LDS A staging:
//   Out[M,N] = X[M,K] * W[N,K]^T   (+ optional softplus epilogue)
// Block = 256 threads (8 waves). Block computes a 64(M) x 128(N) slab:
//   - A slab (64 rows) is shared by all 8 waves -> staged in LDS via
//     global_load_async_to_lds_b128 (1 coalesced b128/thread/chunk = exactly
//     the 64x16 chunk), double-buffered on ASYNCcnt (s_wait_asynccnt 0x1).
//   - Each wave owns one 16-col N tile (4 accumulator chains along M);
//     B fragments are per-wave register-direct loads.
//   - A fragments read back with conflict-free ds_load_b64 (PS=20 padding).
// EPI==1: fused softplus(v + bias[n])  (dt_proj path)
// ---------------------------------------------------------------------------
template <int EPI>
__global__ void gemm_wmma(const float* __restrict__ X, const float* __restrict__ W,
                          const float* __restrict__ bias, float* __restrict__ Out,
                          int N, int K, int lda, int ldb, int ldc) {
    __shared__ float abuf[2][64 * PS];

    const int lane = threadIdx.x & 31;
    const int wv   = threadIdx.x >> 5;               // wave in block: 0..7
    const int nt_n = N >> 4;
    int ni = blockIdx.y * 8 + wv;
    if (ni >= nt_n) ni = nt_n - 1;                   // duplicate work; keeps barriers uniform
    const int m0 = blockIdx.x << 6;
    const int n0 = ni << 4;

    // --- async A staging: thread t moves rows m0+t/4, floats [t%4*4 .. +4) of each chunk
    const int arow = threadIdx.x >> 2;               // 0..63
    const int aseg = (threadIdx.x & 3) << 2;         // 0,4,8,12
    const float*   ag = X + (size_t)(m0 + arow) * lda + aseg;
    const uint32_t l0 = (uint32_t)(uintptr_t)&abuf[0][arow * PS + aseg];
    const uint32_t l1 = (uint32_t)(uintptr_t)&abuf[1][arow * PS + aseg];

    // --- WMMA fragment addressing
    const int kh   = (lane >> 4) << 1;               // K sub-offset: 0 or 2
    const float* wr = W + (size_t)(n0 + (lane & 15)) * ldb + kh;     // B row (N x K)
    const int lrow  = (lane & 15) * PS + kh;         // LDS base index of A frag, tile t adds t*16*PS

    v8f acc[4] = {{0,0,0,0,0,0,0,0}, {0,0,0,0,0,0,0,0},
                  {0,0,0,0,0,0,0,0}, {0,0,0,0,0,0,0,0}};

    // prologue: kick off chunk 0
    asm volatile("global_load_async_to_lds_b128 %0, %1, off" :: "v"(l0), "v"(ag) : "memory");

    const int nchunks = K >> 4;
    for (int c = 0; c < nchunks; ++c) {
        if (c + 1 < nchunks) {                       // prefetch chunk c+1, keep it in flight
            const float*   gnext = ag + (size_t)(c + 1) * KC;
            const uint32_t lnext = ((c + 1) & 1) ? l1 : l0;
            asm volatile("global_load_async_to_lds_b128 %0, %1, off" :: "v"(lnext), "v"(gnext) : "memory");
            asm volatile("s_wait_asynccnt 0x1" ::: "memory");   // chunk c has landed
        } else {
            asm volatile("s_wait_asynccnt 0x0" ::: "memory");
        }
        __syncthreads();                             // chunk c visible block-wide

        const float* ab = abuf[c & 1];
        const int kb = c * KC;
        v2f b[4];
        v2f a[4][4];
#pragma unroll
        for (int j = 0; j < 4; ++j)
            b[j] = *reinterpret_cast<const v2f*>(wr + kb + 4 * j);
#pragma unroll
        for (int t = 0; t < 4; ++t)
#pragma unroll
            for (int j = 0; j < 4; ++j)
                a[t][j] = *reinterpret_cast<const v2f*>(ab + lrow + t * 16 * PS + j * 4);
#pragma unroll
        for (int j = 0; j < 4; ++j) {
#pragma unroll
            for (int t = 0; t < 4; ++t) {
                acc[t] = __builtin_amdgcn_wmma_f32_16x16x4_f32(
                    false, a[t][j], false, b[j], (short)0, acc[t], false, false);
            }
        }
        __syncthreads();                             // done reading buf[c&1] before it is overwritten
    }

    const int ncol = n0 + (lane & 15);
    const int moff = (lane >> 4) << 3;               // 0 or 8
    float bv = 0.f;
    if (EPI == 1) bv = bias[ncol];
#pragma unroll
    for (int t = 0; t < 4; ++t) {
        const int mb = m0 + t * 16 + moff;
#pragma unroll
        for (int r = 0; r < 8; ++r) {
            float v = acc[t][r];
            if (EPI == 1) {                          // softplus(v + b)
                v += bv;
                v = (v > 20.f) ? v : log1pf(__expf(v));
            }
            Out[(size_t)(mb + r) * ldc + ncol] = v;
        }
    }
}

// ---------------------------------------------------------------------------
// Causal depthwise conv (width 4) + bias + SiLU.  x = xz[..., :D_INNER]
// ---------------------------------------------------------------------------
__global__ void conv_silu(const float* __restrict__ xz, const float* __restrict__ cw,
                          const float* __restrict__ cb, float* __restrict__ xc) {
    const int idx = blockIdx.x * blockDim.x + threadIdx.x;   // over ROWS*D_INNER
    const int d = idx & (D_INNER - 1);
    const int l = (idx >> 10) & (SEQ - 1);
    const int b = idx >> 17;
    float acc = cb[d];
#pragma unroll
    for (int j = 0; j < D_CONV; ++j) {
        const int lj = l - (D_CONV - 1) + j;
        if (lj >= 0)
            acc += xz[(size_t)(b * SEQ + lj) * (2 * D_INNER) + d] * cw[d * D_CONV + j];
    }
    xc[idx] = acc / (1.f + __expf(-acc));            // silu
}

// ---------------------------------------------------------------------------
// Selective scan: one thread per (b,d); 16 states in registers.
// b = blockIdx.x >> 2 is scalar -> per-timestep B/C rows of `dbl` are wave-
// uniform and go through the scalar (SMEM) broadcast path.
// Fuses +u*D, silu(z) gate; writes y in-place over delta.
// ---------------------------------------------------------------------------
__global__ void scan_kernel(float* __restrict__ delta_y, const float* __restrict__ dbl,
                            const float* __restrict__ xc, const float* __restrict__ xz,
                            const float* __restrict__ A_log, const float* __restrict__ Dp) {
    const int b = blockIdx.x >> 2;                           // scalar batch index
    const int d = ((blockIdx.x & 3) << 8) + threadIdx.x;     // channel 0..1023

    float A[D_STATE], h[D_STATE];
#pragma unroll
    for (int n = 0; n < D_STATE; ++n) {
        A[n] = -__expf(A_log[d * D_STATE + n]);
        h[n] = 0.f;
    }
    const float Dv = Dp[d];

    for (int t = 0; t < SEQ; ++t) {
        const size_t row = (size_t)(b * SEQ + t);
        const float dt = delta_y[row * D_INNER + d];
        const float u  = xc[row * D_INNER + d];
        const float du = dt * u;
        const float* bc = dbl + row * (DT_RANK + 2 * D_STATE);   // uniform address
        float y = 0.f;
#pragma unroll
        for (int n = 0; n < D_STATE; ++n) {
            h[n] = __expf(dt * A[n]) * h[n] + du * bc[DT_RANK + n];
            y += h[n] * bc[DT_RANK + D_STATE + n];
        }
        y += u * Dv;
        const float z = xz[row * (2 * D_INNER) + D_INNER + d];
        y *= z / (1.f + __expf(-z));                 // * silu(z)
        delta_y[row * D_INNER + d] = y;
    }
}

// ---------------------------------------------------------------------------
// LayerNorm over D_MODEL=512: one wave32 per row, shfl_xor tree reduce.
// ---------------------------------------------------------------------------
__global__ void layernorm_kernel(const float* __restrict__ in, const float* __restrict__ w,
                                 const float* __restrict__ b, float* __restrict__ out) {
    const int wave = (blockIdx.x * blockDim.x + threadIdx.x) >> 5;
    const int lane = threadIdx.x & 31;
    if (wave >= ROWS) return;
    const float* row = in + (size_t)wave * D_MODEL;

    float vals[D_MODEL / 32];
    float s = 0.f;
#pragma unroll
    for (int i = 0; i < D_MODEL / 32; ++i) { vals[i] = row[lane + i * 32]; s += vals[i]; }
#pragma unroll
    for (int off = 16; off > 0; off >>= 1) s += __shfl_xor(s, off, 32);
    const float mean = s * (1.f / D_MODEL);

    float v = 0.f;
#pragma unroll
    for (int i = 0; i < D_MODEL / 32; ++i) { float dd = vals[i] - mean; v += dd * dd; }
#pragma unroll
    for (int off = 16; off > 0; off >>= 1) v += __shfl_xor(v, off, 32);
    const float rstd = rsqrtf(v * (1.f / D_MODEL) + 1e-5f);

    float* orow = out + (size_t)wave * D_MODEL;
#pragma unroll
    for (int i = 0; i < D_MODEL / 32; ++i) {
        const int c = lane + i * 32;
        orow[c] = (vals[i] - mean) * rstd * w[c] + b[c];
    }
}

// ---------------------------------------------------------------------------
extern "C" void kernel_launch(void* const* d_in, const int* in_sizes, int n_in,
                              void* d_out, int out_size, void* d_ws, size_t ws_size,
                              hipStream_t stream) {
    (void)in_sizes; (void)n_in; (void)out_size; (void)ws_size;
    const float* x_in  = (const float*)d_in[0];
    const float* in_w  = (const float*)d_in[1];
    const float* cw    = (const float*)d_in[2];
    const float* cb    = (const float*)d_in[3];
    const float* xpw   = (const float*)d_in[4];
    const float* dtw   = (const float*)d_in[5];
    const float* dtb   = (const float*)d_in[6];
    const float* Alog  = (const float*)d_in[7];
    const float* Dp    = (const float*)d_in[8];
    const float* ow    = (const float*)d_in[9];
    const float* lnw   = (const float*)d_in[10];
    const float* lnb   = (const float*)d_in[11];

    // workspace layout (floats): xz | xc | dbl | delta(->y) | proj | xbuf
    float* ws    = (float*)d_ws;
    float* xz    = ws;                                   // 4096*2048
    float* xc    = xz    + (size_t)ROWS * 2 * D_INNER;   // 4096*1024
    float* dbl   = xc    + (size_t)ROWS * D_INNER;       // 4096*64
    float* delta = dbl   + (size_t)ROWS * (DT_RANK + 2 * D_STATE);  // 4096*1024
    float* proj  = delta + (size_t)ROWS * D_INNER;       // 4096*512
    float* xbuf  = proj  + (size_t)ROWS * D_MODEL;       // 4096*512

    for (int layer = 0; layer < NUM_LAYERS; ++layer) {
        const float* u     = (layer == 0) ? x_in : xbuf;
        float*       lnout = (layer == NUM_LAYERS - 1) ? (float*)d_out : xbuf;

        const float* W1   = in_w + (size_t)layer * 2 * D_INNER * D_MODEL;
        const float* cw_l = cw   + (size_t)layer * D_INNER * D_CONV;
        const float* cb_l = cb   + (size_t)layer * D_INNER;
        const float* xp_l = xpw  + (size_t)layer * (DT_RANK + 2 * D_STATE) * D_INNER;
        const float* dw_l = dtw  + (size_t)layer * D_INNER * DT_RANK;
        const float* db_l = dtb  + (size_t)layer * D_INNER;
        const float* Al_l = Alog + (size_t)layer * D_INNER * D_STATE;
        const float* Dp_l = Dp   + (size_t)layer * D_INNER;
        const float* ow_l = ow   + (size_t)layer * D_MODEL * D_INNER;
        const float* lw_l = lnw  + (size_t)layer * D_MODEL;
        const float* lb_l = lnb  + (size_t)layer * D_MODEL;

        // 1) xz = u @ in_w^T        M=4096 N=2048 K=512  -> grid (M/64, N/128)
        gemm_wmma<0><<<dim3(ROWS / 64, 16), 256, 0, stream>>>(
            u, W1, nullptr, xz, 2 * D_INNER, D_MODEL, D_MODEL, D_MODEL, 2 * D_INNER);

        // 2) depthwise conv + bias + silu
        conv_silu<<<(ROWS * D_INNER) / 256, 256, 0, stream>>>(xz, cw_l, cb_l, xc);

        // 3) dbl = xc @ xpw^T       M=4096 N=64 K=1024   (waves 4..7 duplicate)
        gemm_wmma<0><<<dim3(ROWS / 64, 1), 256, 0, stream>>>(
            xc, xp_l, nullptr, dbl, DT_RANK + 2 * D_STATE, D_INNER,
            D_INNER, D_INNER, DT_RANK + 2 * D_STATE);

        // 4) delta = softplus(dt @ dtw^T + dtb)   M=4096 N=1024 K=32 (X=dbl, lda=64)
        gemm_wmma<1><<<dim3(ROWS / 64, 8), 256, 0, stream>>>(
            dbl, dw_l, db_l, delta, D_INNER, DT_RANK,
            DT_RANK + 2 * D_STATE, DT_RANK, D_INNER);

        // 5) selective scan + D skip + silu(z) gate  (y written over delta)
        scan_kernel<<<BATCH * 4, 256, 0, stream>>>(delta, dbl, xc, xz, Al_l, Dp_l);

        // 6) proj = y @ ow^T        M=4096 N=512 K=1024
        gemm_wmma<0><<<dim3(ROWS / 64, 4), 256, 0, stream>>>(
            delta, ow_l, nullptr, proj, D_MODEL, D_INNER, D_INNER, D_INNER, D_MODEL);

        // 7) layernorm
        layernorm_kernel<<<(ROWS * 32) / 256, 256, 0, stream>>>(proj, lw_l, lb_l, lnout);
    }
}